// GraphNonlinearTerms_39754217292304
// MI455X (gfx1250) — compile-verified
//
#include <hip/hip_runtime.h>

typedef __attribute__((ext_vector_type(2))) float v2f;
typedef __attribute__((ext_vector_type(4))) float v4f;
typedef __attribute__((ext_vector_type(8))) float v8f;

#define BATCH   32768
#define NTILES  (BATCH / 16)   // 2048 tiles of 16 batch rows
#define BLOCKS  256
#define TPB     128            // 4 waves per block

__device__ __forceinline__ v8f wmma_f32_k4(v2f a, v2f b, v8f c) {
  // V_WMMA_F32_16X16X4_F32 : D = A(16x4,f32) * B(4x16,f32) + C(16x16,f32)
  return __builtin_amdgcn_wmma_f32_16x16x4_f32(
      /*neg_a=*/false, a, /*neg_b=*/false, b,
      /*c_mod=*/(short)0, c, /*reuse_a=*/false, /*reuse_b=*/false);
}

__global__ __launch_bounds__(TPB) void gcn_fused_kernel(
    const float* __restrict__ x,
    const float* __restrict__ qW1, const float* __restrict__ qb1,
    const float* __restrict__ qW2, const float* __restrict__ qb2,
    const float* __restrict__ cW1, const float* __restrict__ cb1,
    const float* __restrict__ cW2, const float* __restrict__ cb2,
    const float* __restrict__ eTW1, const float* __restrict__ eTb1,
    const float* __restrict__ eTW2, const float* __restrict__ eTb2,
    const float* __restrict__ eHW1, const float* __restrict__ eHb1,
    const float* __restrict__ eHW2, const float* __restrict__ eHb2,
    float* __restrict__ out)
{
  __shared__ float s_w2m[2][64];      // mean over columns of W2, per net
  __shared__ float s_eW1[2][5][32];   // enso W1, [net][feature][hidden]
  __shared__ float s_eb1[2][32];
  __shared__ float s_eW2[2][32];
  __shared__ float s_eb2[2];

  const int tid = threadIdx.x;

  // ---- cooperative preload: w2m = mean over the 32 output cols of W2 (64 rows x 2 nets) ----
  {
    const int net = tid >> 6;
    const int h   = tid & 63;
    const float* W2 = net ? cW2 : qW2;
    const v4f* p = (const v4f*)(W2 + h * 32);
    float s = 0.0f;
#pragma unroll
    for (int i = 0; i < 8; ++i) { v4f v = p[i]; s += v.x + v.y + v.z + v.w; }
    s_w2m[net][h] = s * (1.0f / 32.0f);
  }
  // ---- enso MLP weights into LDS ----
  if (tid < 32) {
#pragma unroll
    for (int f = 0; f < 5; ++f) {
      s_eW1[0][f][tid] = eTW1[f * 32 + tid];
      s_eW1[1][f][tid] = eHW1[f * 32 + tid];
    }
    s_eb1[0][tid] = eTb1[tid];
    s_eb1[1][tid] = eHb1[tid];
    s_eW2[0][tid] = eTW2[tid];
    s_eW2[1][tid] = eHW2[tid];
    if (tid < 2) s_eb2[tid] = tid ? eHb2[0] : eTb2[0];
  }
  __syncthreads();

  const int lane = tid & 31;
  const int half = lane >> 4;     // 0: lanes 0-15, 1: lanes 16-31
  const int lm   = lane & 15;
  const int koff = half * 2;      // K-pair owned by this lane half (f32 16x16x4 layout)

  // ---- loop-invariant per-lane scalars: b1/w2m per (net, N-tile), mean(b2q)+mean(b2c) ----
  float b1r[2][4], w2mr[2][4];
#pragma unroll
  for (int n = 0; n < 4; ++n) {
    const int col = n * 16 + lm;              // N index = HID column
    b1r[0][n]  = qb1[col];
    b1r[1][n]  = cb1[col];
    w2mr[0][n] = s_w2m[0][col];
    w2mr[1][n] = s_w2m[1][col];
  }
  float mb2 = qb2[lane] + cb2[lane];
#pragma unroll
  for (int m = 16; m >= 1; m >>= 1) mb2 += __shfl_xor(mb2, m, 32);
  mb2 *= (1.0f / 32.0f);

  // ---- preload B fragments: W1 (32x64) for both nets, kept in VGPRs ----
  // B(4x16) lane layout: lanes 0-15 hold K rows {0,1}, lanes 16-31 hold K rows {2,3}, N = lane%16.
  v2f bq[4][8], bc[4][8];
#pragma unroll
  for (int n = 0; n < 4; ++n) {
    const int col = n * 16 + lm;
#pragma unroll
    for (int k = 0; k < 8; ++k) {
      const int kr = k * 4 + koff;
      v2f t;
      t.x = qW1[kr * 64 + col]; t.y = qW1[(kr + 1) * 64 + col]; bq[n][k] = t;
      t.x = cW1[kr * 64 + col]; t.y = cW1[(kr + 1) * 64 + col]; bc[n][k] = t;
    }
  }

  const int wgid  = blockIdx.x * (TPB >> 5) + (tid >> 5);
  const int wstep = gridDim.x * (TPB >> 5);

  for (int t = wgid; t < NTILES; t += wstep) {
    const int base = t * 16;

    // ---- A fragments: 16 batch rows x 32 features of x ----
    // A(16x4) lane layout: row M = lane%16; lanes 0-15 hold K {0,1}, lanes 16-31 hold K {2,3}.
    v2f a[8];
    const float* xr = x + (base + lm) * 32 + koff;
#pragma unroll
    for (int k = 0; k < 8; ++k) a[k] = *(const v2f*)(xr + 4 * k);

    float accv[8];
#pragma unroll
    for (int v = 0; v < 8; ++v) accv[v] = 0.0f;

    // ---- quad net: C = X(16x32) @ W1(32x64), folded with relu + w2m dot ----
#pragma unroll
    for (int n = 0; n < 4; ++n) {
      v8f c = {0.f, 0.f, 0.f, 0.f, 0.f, 0.f, 0.f, 0.f};
#pragma unroll
      for (int k = 0; k < 8; ++k) c = wmma_f32_k4(a[k], bq[n][k], c);
#pragma unroll
      for (int v = 0; v < 8; ++v) {
        // C element: row M = v + 8*half, col N = n*16 + lane%16
        const float h = fmaxf(c[v] + b1r[0][n], 0.0f);
        accv[v] = fmaf(h, w2mr[0][n], accv[v]);
      }
    }
    // ---- cubic net ----
#pragma unroll
    for (int n = 0; n < 4; ++n) {
      v8f c = {0.f, 0.f, 0.f, 0.f, 0.f, 0.f, 0.f, 0.f};
#pragma unroll
      for (int k = 0; k < 8; ++k) c = wmma_f32_k4(a[k], bc[n][k], c);
#pragma unroll
      for (int v = 0; v < 8; ++v) {
        const float h = fmaxf(c[v] + b1r[1][n], 0.0f);
        accv[v] = fmaf(h, w2mr[1][n], accv[v]);
      }
    }

    // ---- reduce over N (the 16 lanes of each half) -> nl scalar per batch row ----
#pragma unroll
    for (int v = 0; v < 8; ++v) {
      float s = accv[v];
#pragma unroll
      for (int m = 8; m >= 1; m >>= 1) s += __shfl_xor(s, m, 32);
      accv[v] = s + mb2;   // nl for row (v + 8*half); identical across the half's lanes
    }

    // ---- enso 5->32->1 MLPs: lane = (row = lane%16, net = half) ----
    const float T = x[(base + lm) * 32 + 0];
    const float H = x[(base + lm) * 32 + 1];
    const float f0 = T, f1 = H, f2 = T * T, f3 = T * H;
    const float f4 = half ? (T * H * H) : (T * T * T);
    const int net = half;
    float cacc = s_eb2[net];
#pragma unroll 4
    for (int j = 0; j < 32; ++j) {
      float hj = s_eb1[net][j];
      hj = fmaf(f0, s_eW1[net][0][j], hj);
      hj = fmaf(f1, s_eW1[net][1][j], hj);
      hj = fmaf(f2, s_eW1[net][2][j], hj);
      hj = fmaf(f3, s_eW1[net][3][j], hj);
      hj = fmaf(f4, s_eW1[net][4][j], hj);
      cacc = fmaf(fmaxf(hj, 0.0f), s_eW2[net][j], cacc);
    }
    // lane r (r<16) holds cT[row r]; lane r+16 holds cH[row r]

    // ---- write 16 rows x 32 cols, coalesced (lane = output column) ----
#pragma unroll
    for (int r = 0; r < 16; ++r) {
      const float nlr = __shfl(accv[r & 7], (r < 8) ? 0 : 16, 32);
      const float cTr = __shfl(cacc, r, 32);
      const float cHr = __shfl(cacc, r + 16, 32);
      float val = nlr;
      if (lane == 0) val += cTr;
      if (lane == 1) val += cHr;
      out[(base + r) * 32 + lane] = val;
    }
  }
}

extern "C" void kernel_launch(void* const* d_in, const int* in_sizes, int n_in,
                              void* d_out, int out_size, void* d_ws, size_t ws_size,
                              hipStream_t stream) {
  (void)in_sizes; (void)n_in; (void)out_size; (void)d_ws; (void)ws_size;
  const float* x    = (const float*)d_in[0];
  const float* qW1  = (const float*)d_in[1];
  const float* qb1  = (const float*)d_in[2];
  const float* qW2  = (const float*)d_in[3];
  const float* qb2  = (const float*)d_in[4];
  const float* cW1  = (const float*)d_in[5];
  const float* cb1  = (const float*)d_in[6];
  const float* cW2  = (const float*)d_in[7];
  const float* cb2  = (const float*)d_in[8];
  const float* eTW1 = (const float*)d_in[9];
  const float* eTb1 = (const float*)d_in[10];
  const float* eTW2 = (const float*)d_in[11];
  const float* eTb2 = (const float*)d_in[12];
  const float* eHW1 = (const float*)d_in[13];
  const float* eHb1 = (const float*)d_in[14];
  const float* eHW2 = (const float*)d_in[15];
  const float* eHb2 = (const float*)d_in[16];
  float* out = (float*)d_out;

  gcn_fused_kernel<<<BLOCKS, TPB, 0, stream>>>(
      x, qW1, qb1, qW2, qb2, cW1, cb1, cW2, cb2,
      eTW1, eTb1, eTW2, eTb2, eHW1, eHb1, eHW2, eHb2, out);
}